// HeteroGATLayer_12678743458335
// MI455X (gfx1250) — compile-verified
//
#include <hip/hip_runtime.h>
#include <float.h>

// Problem constants (match reference)
#define NB 100000
#define NG 20000
#define E1 400000
#define E2 40000
// F=128, FE=16, H=4, D=32, OUT=128

typedef __attribute__((ext_vector_type(2))) float v2f;
typedef __attribute__((ext_vector_type(8))) float v8f;

// ---------------------------------------------------------------------------
// fill
__global__ void fill_k(float* __restrict__ p, size_t n, float v) {
    size_t i = (size_t)blockIdx.x * blockDim.x + threadIdx.x;
    if (i < n) p[i] = v;
}

// ---------------------------------------------------------------------------
// wa3[f*4+h] = sum_d W_edge[f,128][h*32+d] * a_line[64+d]; ba3[h] likewise for bias
__global__ void wa3_k(const float* __restrict__ We, const float* __restrict__ be,
                      const float* __restrict__ aL, float* __restrict__ wa3,
                      float* __restrict__ ba3) {
    int t = threadIdx.x;
    if (t < 64) {
        int f = t >> 2, h = t & 3;
        float s = 0.f;
        for (int d = 0; d < 32; ++d) s += We[f * 128 + h * 32 + d] * aL[64 + d];
        wa3[t] = s;
    } else if (t < 68) {
        int h = t - 64;
        float s = 0.f;
        for (int d = 0; d < 32; ++d) s += be[h * 32 + d] * aL[64 + d];
        ba3[h] = s;
    }
}

// ---------------------------------------------------------------------------
// D[M,128] = A[M,128] @ W[128,128] + bias ; optional relu.
// One wave per 16-row x 64-col tile-group (4 WMMA col tiles, shared A frag).
__global__ __launch_bounds__(256) void gemm128_wmma(
    const float* __restrict__ A, const float* __restrict__ W,
    const float* __restrict__ bias, float* __restrict__ D, int M, int relu) {
    const int wave = threadIdx.x >> 5;
    const int lane = threadIdx.x & 31;
    const int tile = blockIdx.x * 8 + wave;
    const int ntiles = M >> 4;
    if (tile >= ntiles) return;           // uniform per wave: EXEC stays all-1s
    const int half = lane >> 4;
    const int l16 = lane & 15;
    const int row0 = tile << 4;
    const int cb = blockIdx.y << 6;

    const float* Ar = A + (size_t)(row0 + l16) * 128 + 2 * half;
    v8f acc0 = {}, acc1 = {}, acc2 = {}, acc3 = {};
#pragma unroll 4
    for (int k = 0; k < 128; k += 4) {
        v2f a; a.x = Ar[k]; a.y = Ar[k + 1];
        const float* Wk = W + (size_t)(k + 2 * half) * 128 + cb + l16;
        v2f b0, b1, b2, b3;
        b0.x = Wk[0];  b0.y = Wk[128];
        b1.x = Wk[16]; b1.y = Wk[144];
        b2.x = Wk[32]; b2.y = Wk[160];
        b3.x = Wk[48]; b3.y = Wk[176];
        acc0 = __builtin_amdgcn_wmma_f32_16x16x4_f32(false, a, false, b0, (short)0, acc0, false, false);
        acc1 = __builtin_amdgcn_wmma_f32_16x16x4_f32(false, a, false, b1, (short)0, acc1, false, false);
        acc2 = __builtin_amdgcn_wmma_f32_16x16x4_f32(false, a, false, b2, (short)0, acc2, false, false);
        acc3 = __builtin_amdgcn_wmma_f32_16x16x4_f32(false, a, false, b3, (short)0, acc3, false, false);
    }
    const float bs0 = bias[cb + l16], bs1 = bias[cb + 16 + l16];
    const float bs2 = bias[cb + 32 + l16], bs3 = bias[cb + 48 + l16];
#pragma unroll
    for (int v = 0; v < 8; ++v) {
        int row = row0 + v + 8 * half;
        float* Dr = D + (size_t)row * 128 + cb + l16;
        float x0 = acc0[v] + bs0, x1 = acc1[v] + bs1;
        float x2 = acc2[v] + bs2, x3 = acc3[v] + bs3;
        if (relu) {
            x0 = fmaxf(x0, 0.f); x1 = fmaxf(x1, 0.f);
            x2 = fmaxf(x2, 0.f); x3 = fmaxf(x3, 0.f);
        }
        Dr[0] = x0; Dr[16] = x1; Dr[32] = x2; Dr[48] = x3;
    }
}

// ---------------------------------------------------------------------------
// Update MLP: D = relu(concat_headwise(P, AG) @ W[256,128] + bias)
// Flattened concat layout per node: k -> head h=k>>6, w=k&63; w<32 from P else AG.
// Aligned K pairs never straddle the 32-boundary, so pick base per pair.
__global__ __launch_bounds__(256) void gemm_upd_wmma(
    const float* __restrict__ P, const float* __restrict__ AG,
    const float* __restrict__ W, const float* __restrict__ bias,
    float* __restrict__ D, int M) {
    const int wave = threadIdx.x >> 5;
    const int lane = threadIdx.x & 31;
    const int tile = blockIdx.x * 8 + wave;
    const int ntiles = M >> 4;
    if (tile >= ntiles) return;
    const int half = lane >> 4;
    const int l16 = lane & 15;
    const int row0 = tile << 4;
    const int cb = blockIdx.y << 6;

    const size_t rb = (size_t)(row0 + l16) * 128;
    v8f acc0 = {}, acc1 = {}, acc2 = {}, acc3 = {};
#pragma unroll 4
    for (int k = 0; k < 256; k += 4) {
        int kk = k + 2 * half;
        int h = kk >> 6, w = kk & 63;
        const float* src = (w < 32) ? P : AG;
        const float* Ap = src + rb + h * 32 + (w & 31);
        v2f a; a.x = Ap[0]; a.y = Ap[1];
        const float* Wk = W + (size_t)kk * 128 + cb + l16;
        v2f b0, b1, b2, b3;
        b0.x = Wk[0];  b0.y = Wk[128];
        b1.x = Wk[16]; b1.y = Wk[144];
        b2.x = Wk[32]; b2.y = Wk[160];
        b3.x = Wk[48]; b3.y = Wk[176];
        acc0 = __builtin_amdgcn_wmma_f32_16x16x4_f32(false, a, false, b0, (short)0, acc0, false, false);
        acc1 = __builtin_amdgcn_wmma_f32_16x16x4_f32(false, a, false, b1, (short)0, acc1, false, false);
        acc2 = __builtin_amdgcn_wmma_f32_16x16x4_f32(false, a, false, b2, (short)0, acc2, false, false);
        acc3 = __builtin_amdgcn_wmma_f32_16x16x4_f32(false, a, false, b3, (short)0, acc3, false, false);
    }
    const float bs0 = bias[cb + l16], bs1 = bias[cb + 16 + l16];
    const float bs2 = bias[cb + 32 + l16], bs3 = bias[cb + 48 + l16];
#pragma unroll
    for (int v = 0; v < 8; ++v) {
        int row = row0 + v + 8 * half;
        float* Dr = D + (size_t)row * 128 + cb + l16;
        Dr[0]  = fmaxf(acc0[v] + bs0, 0.f);
        Dr[16] = fmaxf(acc1[v] + bs1, 0.f);
        Dr[32] = fmaxf(acc2[v] + bs2, 0.f);
        Dr[48] = fmaxf(acc3[v] + bs3, 0.f);
    }
}

// ---------------------------------------------------------------------------
// Per-bus-node score scalars: sa13_line=(a1+a3)·p, da2_line=a2·p, sa13_feeds=(a1f+a3f)·p
__global__ void node_pre_bus_k(const float* __restrict__ p, const float* __restrict__ aL,
                               const float* __restrict__ aF, float* __restrict__ sa13L,
                               float* __restrict__ da2L, float* __restrict__ sa13F) {
    int idx = blockIdx.x * blockDim.x + threadIdx.x;
    if (idx >= NB * 4) return;
    int n = idx >> 2, h = idx & 3;
    const float* v = p + (size_t)n * 128 + h * 32;
    float s1 = 0.f, s2 = 0.f, s3 = 0.f;
#pragma unroll
    for (int d = 0; d < 32; ++d) {
        float x = v[d];
        s1 += x * (aL[d] + aL[64 + d]);
        s2 += x * aL[32 + d];
        s3 += x * (aF[d] + aF[64 + d]);
    }
    sa13L[idx] = s1; da2L[idx] = s2; sa13F[idx] = s3;
}

__global__ void node_pre_gen_k(const float* __restrict__ p, const float* __restrict__ aF,
                               float* __restrict__ da2F) {
    int idx = blockIdx.x * blockDim.x + threadIdx.x;
    if (idx >= NG * 4) return;
    int n = idx >> 2, h = idx & 3;
    const float* v = p + (size_t)n * 128 + h * 32;
    float s2 = 0.f;
#pragma unroll
    for (int d = 0; d < 32; ++d) s2 += v[d] * aF[32 + d];
    da2F[idx] = s2;
}

// ---------------------------------------------------------------------------
// Line-edge scores: leaky_relu(sa13[s]+da2[r]+ba3+ef·wa3); atomic segment max
__global__ void scores_line_k(const int* __restrict__ se, const int* __restrict__ re,
                              const float* __restrict__ ef, const float* __restrict__ sa13,
                              const float* __restrict__ da2, const float* __restrict__ wa3,
                              const float* __restrict__ ba3, float* __restrict__ sc,
                              float* __restrict__ segmax) {
    int idx = blockIdx.x * blockDim.x + threadIdx.x;
    if (idx >= E1 * 4) return;
    int e = idx >> 2, h = idx & 3;
    int s = se[e], r = re[e];
    float v = sa13[s * 4 + h] + da2[r * 4 + h] + ba3[h];
    const float* efe = ef + (size_t)e * 16;
#pragma unroll
    for (int f = 0; f < 16; ++f) v += efe[f] * wa3[f * 4 + h];
    v = v > 0.f ? v : 0.2f * v;
    sc[idx] = v;
    atomicMax(&segmax[r * 4 + h], v);
}

__global__ void scores_feeds_k(const int* __restrict__ se, const int* __restrict__ re,
                               const float* __restrict__ sa13, const float* __restrict__ da2,
                               float* __restrict__ sc, float* __restrict__ segmax) {
    int idx = blockIdx.x * blockDim.x + threadIdx.x;
    if (idx >= E2 * 4) return;
    int e = idx >> 2, h = idx & 3;
    int s = se[e], r = re[e];
    float v = sa13[s * 4 + h] + da2[r * 4 + h];
    v = v > 0.f ? v : 0.2f * v;
    sc[idx] = v;
    atomicMax(&segmax[r * 4 + h], v);
}

// exp(score - segmax[recv]) in place; atomic segment sum
__global__ void expsum_k(float* __restrict__ sc, const int* __restrict__ re,
                         const float* __restrict__ segmax, float* __restrict__ segsum, int E) {
    int idx = blockIdx.x * blockDim.x + threadIdx.x;
    if (idx >= E * 4) return;
    int e = idx >> 2, h = idx & 3;
    int r = re[e];
    float v = __expf(sc[idx] - segmax[r * 4 + h]);
    sc[idx] = v;
    atomicAdd(&segsum[r * 4 + h], v);
}

// one wave per edge: agg[r] += p_src[s] * alpha  (lane*4 floats each)
__global__ __launch_bounds__(256) void scatter_k(
    const float* __restrict__ sc, const int* __restrict__ se, const int* __restrict__ re,
    const float* __restrict__ segsum, const float* __restrict__ psrc,
    float* __restrict__ agg, int E) {
    int wave = threadIdx.x >> 5, lane = threadIdx.x & 31;
    int e = blockIdx.x * 8 + wave;
    if (e >= E) return;
    int s = se[e], r = re[e];
    int h = lane >> 3;
    float alpha = sc[e * 4 + h] / (segsum[r * 4 + h] + 1e-8f);
    const float4 v = *(const float4*)(psrc + (size_t)s * 128 + lane * 4);
    float* d = agg + (size_t)r * 128 + lane * 4;
    atomicAdd(d + 0, v.x * alpha);
    atomicAdd(d + 1, v.y * alpha);
    atomicAdd(d + 2, v.z * alpha);
    atomicAdd(d + 3, v.w * alpha);
}

// ---------------------------------------------------------------------------
extern "C" void kernel_launch(void* const* d_in, const int* in_sizes, int n_in,
                              void* d_out, int out_size, void* d_ws, size_t ws_size,
                              hipStream_t stream) {
    (void)in_sizes; (void)n_in; (void)out_size; (void)ws_size;
    const float* bus_x       = (const float*)d_in[0];
    const float* gen_x       = (const float*)d_in[1];
    const float* ef_line     = (const float*)d_in[2];
    const float* w_proj_bus  = (const float*)d_in[3];
    const float* b_proj_bus  = (const float*)d_in[4];
    const float* w_proj_gen  = (const float*)d_in[5];
    const float* b_proj_gen  = (const float*)d_in[6];
    const float* w_edge_line = (const float*)d_in[7];
    const float* b_edge_line = (const float*)d_in[8];
    const float* a_line      = (const float*)d_in[9];
    const float* a_feeds     = (const float*)d_in[10];
    const float* w_upd_bus   = (const float*)d_in[11];
    const float* b_upd_bus   = (const float*)d_in[12];
    const float* w_upd_gen   = (const float*)d_in[13];
    const float* b_upd_gen   = (const float*)d_in[14];
    const int* senders_line    = (const int*)d_in[15];
    const int* receivers_line  = (const int*)d_in[16];
    const int* senders_feeds   = (const int*)d_in[17];
    const int* receivers_feeds = (const int*)d_in[18];
    float* out = (float*)d_out;

    // workspace layout (floats)
    float* ws = (float*)d_ws;
    size_t o = 0;
    float* p_bus      = ws + o; o += (size_t)NB * 128;
    float* p_gen      = ws + o; o += (size_t)NG * 128;
    float* agg_bus    = ws + o; o += (size_t)NB * 128;   // zero region start
    float* agg_gen    = ws + o; o += (size_t)NG * 128;
    float* segsum_bus = ws + o; o += (size_t)NB * 4;
    float* segsum_gen = ws + o; o += (size_t)NG * 4;     // zero region end
    float* segmax_bus = ws + o; o += (size_t)NB * 4;     // -inf region start
    float* segmax_gen = ws + o; o += (size_t)NG * 4;     // -inf region end
    float* sa13_line  = ws + o; o += (size_t)NB * 4;
    float* da2_line   = ws + o; o += (size_t)NB * 4;
    float* sa13_feeds = ws + o; o += (size_t)NB * 4;
    float* da2_feeds  = ws + o; o += (size_t)NG * 4;
    float* sc_line    = ws + o; o += (size_t)E1 * 4;
    float* sc_feeds   = ws + o; o += (size_t)E2 * 4;
    float* wa3        = ws + o; o += 64;
    float* ba3        = ws + o; o += 4;

    // 1) init accumulators
    size_t nz = (size_t)(NB + NG) * 128 + (size_t)(NB + NG) * 4;
    fill_k<<<(unsigned)((nz + 255) / 256), 256, 0, stream>>>(agg_bus, nz, 0.f);
    size_t nm = (size_t)(NB + NG) * 4;
    fill_k<<<(unsigned)((nm + 255) / 256), 256, 0, stream>>>(segmax_bus, nm, -FLT_MAX);

    // 2) tiny precompute for folded edge projection
    wa3_k<<<1, 128, 0, stream>>>(w_edge_line, b_edge_line, a_line, wa3, ba3);

    // 3) node projections (WMMA f32)
    gemm128_wmma<<<dim3((NB / 16 + 7) / 8, 2), 256, 0, stream>>>(
        bus_x, w_proj_bus, b_proj_bus, p_bus, NB, 0);
    gemm128_wmma<<<dim3((NG / 16 + 7) / 8, 2), 256, 0, stream>>>(
        gen_x, w_proj_gen, b_proj_gen, p_gen, NG, 0);

    // 4) per-node attention scalars
    node_pre_bus_k<<<(NB * 4 + 255) / 256, 256, 0, stream>>>(
        p_bus, a_line, a_feeds, sa13_line, da2_line, sa13_feeds);
    node_pre_gen_k<<<(NG * 4 + 255) / 256, 256, 0, stream>>>(p_gen, a_feeds, da2_feeds);

    // 5) scores + segment max
    scores_line_k<<<(E1 * 4 + 255) / 256, 256, 0, stream>>>(
        senders_line, receivers_line, ef_line, sa13_line, da2_line, wa3, ba3,
        sc_line, segmax_bus);
    scores_feeds_k<<<(E2 * 4 + 255) / 256, 256, 0, stream>>>(
        senders_feeds, receivers_feeds, sa13_feeds, da2_feeds, sc_feeds, segmax_gen);

    // 6) exp + segment sum
    expsum_k<<<(E1 * 4 + 255) / 256, 256, 0, stream>>>(
        sc_line, receivers_line, segmax_bus, segsum_bus, E1);
    expsum_k<<<(E2 * 4 + 255) / 256, 256, 0, stream>>>(
        sc_feeds, receivers_feeds, segmax_gen, segsum_gen, E2);

    // 7) weighted message scatter (one wave per edge)
    scatter_k<<<E1 / 8, 256, 0, stream>>>(
        sc_line, senders_line, receivers_line, segsum_bus, p_bus, agg_bus, E1);
    scatter_k<<<E2 / 8, 256, 0, stream>>>(
        sc_feeds, senders_feeds, receivers_feeds, segsum_gen, p_gen, agg_gen, E2);

    // 8) update MLPs -> output [NB+NG, 128]
    gemm_upd_wmma<<<dim3((NB / 16 + 7) / 8, 2), 256, 0, stream>>>(
        p_bus, agg_bus, w_upd_bus, b_upd_bus, out, NB);
    gemm_upd_wmma<<<dim3((NG / 16 + 7) / 8, 2), 256, 0, stream>>>(
        p_gen, agg_gen, w_upd_gen, b_upd_gen, out + (size_t)NB * 128, NG);
}